// LengthRegulator_54228257079707
// MI455X (gfx1250) — compile-verified
//
#include <hip/hip_runtime.h>
#include <hip/hip_bf16.h>

#define T_LEN 512
#define D_DIM 384
#define WPB   8   // waves per block in expand kernel (wave32)

typedef float v4f __attribute__((ext_vector_type(4)));
typedef int   v4i __attribute__((ext_vector_type(4)));

// ---------------------------------------------------------------------------
// gfx1250 async global<->LDS path (ASYNCcnt-tracked DMA-style copies).
// Probe result: signature is (AS1 v4i*, AS3 v4i*, imm offset, imm cpol).
// ---------------------------------------------------------------------------
#if __has_builtin(__builtin_amdgcn_global_load_async_to_lds_b128) && \
    __has_builtin(__builtin_amdgcn_global_store_async_from_lds_b128)
#define HAVE_ASYNC_LDS 1
#else
#define HAVE_ASYNC_LDS 0
#endif

__device__ __forceinline__ void wait_async0() {
#if __has_builtin(__builtin_amdgcn_s_wait_asynccnt)
    __builtin_amdgcn_s_wait_asynccnt(0);
#else
    asm volatile("s_wait_asynccnt 0" ::: "memory");
#endif
}

#if HAVE_ASYNC_LDS
typedef __attribute__((address_space(1))) v4i g_v4i;
typedef __attribute__((address_space(3))) v4i l_v4i;

__device__ __forceinline__ g_v4i* glob_cast(const void* p) {
    return (g_v4i*)(unsigned long long)p;
}
__device__ __forceinline__ l_v4i* lds_cast(void* p) {
    return (l_v4i*)(unsigned int)(unsigned long long)p;
}
#endif

// ---------------------------------------------------------------------------
// Phase 1: per-batch inclusive scan of durations (Hillis-Steele in LDS).
// Writes cum[b][t] (int) and mel_len (int to ws, float to output tail).
// ---------------------------------------------------------------------------
__global__ void lr_scan_kernel(const int* __restrict__ duration,
                               const int* __restrict__ alpha_p,
                               int*       __restrict__ cum,
                               int*       __restrict__ mel_i,
                               float*     __restrict__ mel_out)
{
    __shared__ int s[T_LEN];
    const int b = blockIdx.x;
    const int t = threadIdx.x;

    const int alpha = alpha_p[0];
    int d = duration[b * T_LEN + t] * alpha;   // alpha is an integer scalar
    d = d < 0 ? 0 : d;
    s[t] = d;
    __syncthreads();

    #pragma unroll
    for (int off = 1; off < T_LEN; off <<= 1) {
        int v   = s[t];
        int add = (t >= off) ? s[t - off] : 0;
        __syncthreads();
        s[t] = v + add;
        __syncthreads();
    }

    cum[b * T_LEN + t] = s[t];
    if (t == T_LEN - 1) {
        mel_i[b]   = s[t];
        mel_out[b] = (float)s[t];
    }
}

// ---------------------------------------------------------------------------
// Phase 2: expansion. One wave per output frame: binary search cum (in LDS),
// then stream the 384-float row through LDS with async b128 copies.
// ---------------------------------------------------------------------------
__global__ void lr_expand_kernel(const float* __restrict__ x,
                                 const int*   __restrict__ cum,
                                 const int*   __restrict__ mel_i,
                                 float*       __restrict__ out,
                                 int maxlen)
{
    __shared__ int cum_s[T_LEN];
    __shared__ __align__(16) float stage[WPB][D_DIM];

    const int b = blockIdx.y;

    for (int i = threadIdx.x; i < T_LEN; i += blockDim.x)
        cum_s[i] = cum[b * T_LEN + i];
    __syncthreads();

    const int mel  = mel_i[b];
    const int wave = threadIdx.x >> 5;
    const int lane = threadIdx.x & 31;

    for (int pos = blockIdx.x * WPB + wave; pos < maxlen;
         pos += gridDim.x * WPB) {

        float* dst = out + ((size_t)b * maxlen + pos) * D_DIM;

        if (pos < mel) {
            // searchsorted(cum, pos, side='right'): first i with cum[i] > pos
            int lo = 0, hi = T_LEN;
            while (lo < hi) {
                int mid = (lo + hi) >> 1;
                if (cum_s[mid] <= pos) lo = mid + 1; else hi = mid;
            }
            const int idx = lo < T_LEN ? lo : T_LEN - 1;
            const float* src = x + ((size_t)b * T_LEN + idx) * D_DIM;

#if HAVE_ASYNC_LDS
            float* sw = stage[wave];
            #pragma unroll
            for (int k = 0; k < 3; ++k) {
                const int off = (k * 32 + lane) * 4;
                __builtin_amdgcn_global_load_async_to_lds_b128(
                    glob_cast(src + off), lds_cast(sw + off), 0, 0);
            }
            wait_async0();   // LDS staging complete
            #pragma unroll
            for (int k = 0; k < 3; ++k) {
                const int off = (k * 32 + lane) * 4;
                __builtin_amdgcn_global_store_async_from_lds_b128(
                    glob_cast(dst + off), lds_cast(sw + off), 0, 0);
            }
            wait_async0();   // LDS readable again before next iteration
#else
            #pragma unroll
            for (int k = 0; k < 3; ++k) {
                const int off = (k * 32 + lane) * 4;
                v4f v = *(const v4f*)(src + off);
                __builtin_nontemporal_store(v, (v4f*)(dst + off));
            }
#endif
        } else {
            const v4f z = {0.f, 0.f, 0.f, 0.f};
            #pragma unroll
            for (int k = 0; k < 3; ++k) {
                const int off = (k * 32 + lane) * 4;
                __builtin_nontemporal_store(z, (v4f*)(dst + off));
            }
        }
    }
}

// ---------------------------------------------------------------------------
// Launch
// ---------------------------------------------------------------------------
extern "C" void kernel_launch(void* const* d_in, const int* in_sizes, int n_in,
                              void* d_out, int out_size, void* d_ws, size_t ws_size,
                              hipStream_t stream)
{
    const int B = 16, D = D_DIM;

    const float* x        = (const float*)d_in[0];
    const int*   duration = (const int*)d_in[1];
    const int*   alpha    = (const int*)d_in[2];
    // d_in[3] is max_len (device scalar); derive it host-side from out_size
    // since graph capture forbids synchronous readback.
    const int maxlen = out_size / (B * D);      // = 3584

    float* out     = (float*)d_out;
    float* mel_out = out + (size_t)B * maxlen * D;   // tuple tail: mel_len

    int* cum   = (int*)d_ws;        // [B][T_LEN] ints = 32 KB
    int* mel_i = cum + B * T_LEN;   // [B]

    lr_scan_kernel<<<dim3(B), dim3(T_LEN), 0, stream>>>(duration, alpha,
                                                        cum, mel_i, mel_out);

    dim3 grid((maxlen + WPB - 1) / WPB, B);
    lr_expand_kernel<<<grid, dim3(32 * WPB), 0, stream>>>(x, cum, mel_i,
                                                          out, maxlen);
}